// EmbeddingAverager_7017976562185
// MI455X (gfx1250) — compile-verified
//
#include <hip/hip_runtime.h>

typedef float v2f __attribute__((ext_vector_type(2)));
typedef float v8f __attribute__((ext_vector_type(8)));

constexpr int NUM_IDS = 8192;
constexpr int D       = 512;
constexpr int SEGS_PER_BLOCK = 16;   // one WMMA M-tile of ids per block
constexpr int AGG_ELEMS = NUM_IDS * D;

// ---------------------------------------------------------------------------
// K0: init output (agg=0, uniq=-1) and counts=0
// ---------------------------------------------------------------------------
__global__ void ea_init_kernel(float* __restrict__ out, int* __restrict__ counts) {
  int i = blockIdx.x * blockDim.x + threadIdx.x;
  if (i < AGG_ELEMS) {
    out[i] = 0.0f;
  } else if (i < AGG_ELEMS + NUM_IDS) {
    out[i] = -1.0f;                       // uniq_ids fill_value
  }
  if (i < NUM_IDS) counts[i] = 0;
}

// ---------------------------------------------------------------------------
// K1: histogram of labels
// ---------------------------------------------------------------------------
__global__ void ea_hist_kernel(const int* __restrict__ labels, int n,
                               int* __restrict__ counts) {
  int i = blockIdx.x * blockDim.x + threadIdx.x;
  if (i < n) atomicAdd(&counts[labels[i]], 1);
}

// ---------------------------------------------------------------------------
// K2: single-block dual exclusive scan over 8192 ids:
//   offsets = exscan(counts)          (CSR offsets, cursor copy)
//   rank    = exscan(counts > 0)      (compacted position of each present id)
//   uniq_out[rank[g]] = g             (sorted unique ids)
// ---------------------------------------------------------------------------
__global__ void ea_scan_kernel(const int* __restrict__ counts,
                               int* __restrict__ offsets,
                               int* __restrict__ cursor,
                               int* __restrict__ rank,
                               float* __restrict__ uniq_out,
                               int n_total) {
  __shared__ int s_cnt[256];
  __shared__ int s_prs[256];
  const int t = threadIdx.x;
  const int base = t * (NUM_IDS / 256);   // 32 ids per thread

  int csum = 0, psum = 0;
  for (int k = 0; k < NUM_IDS / 256; ++k) {
    int c = counts[base + k];
    csum += c;
    psum += (c > 0) ? 1 : 0;
  }
  s_cnt[t] = csum;
  s_prs[t] = psum;
  __syncthreads();

  // Hillis-Steele inclusive scan over 256 thread totals
  for (int off = 1; off < 256; off <<= 1) {
    int vc = 0, vp = 0;
    if (t >= off) { vc = s_cnt[t - off]; vp = s_prs[t - off]; }
    __syncthreads();
    if (t >= off) { s_cnt[t] += vc; s_prs[t] += vp; }
    __syncthreads();
  }

  int rc = (t == 0) ? 0 : s_cnt[t - 1];   // exclusive thread base
  int rp = (t == 0) ? 0 : s_prs[t - 1];
  for (int k = 0; k < NUM_IDS / 256; ++k) {
    int g = base + k;
    int c = counts[g];
    offsets[g] = rc;
    cursor[g]  = rc;
    rank[g]    = rp;
    if (c > 0) uniq_out[rp] = (float)g;
    rc += c;
    rp += (c > 0) ? 1 : 0;
  }
  if (t == 0) offsets[NUM_IDS] = n_total;
}

// ---------------------------------------------------------------------------
// K3: counting-sort scatter -> order[] holds sample indices grouped by id
// ---------------------------------------------------------------------------
__global__ void ea_scatter_kernel(const int* __restrict__ labels, int n,
                                  int* __restrict__ cursor,
                                  int* __restrict__ order) {
  int i = blockIdx.x * blockDim.x + threadIdx.x;
  if (i < n) {
    int p = atomicAdd(&cursor[labels[i]], 1);
    order[p] = i;
  }
}

// ---------------------------------------------------------------------------
// K4: WMMA segmented reduction.
// Block b owns ids [b*16, b*16+16).  8 waves/block; wave w owns columns
// [w*64, w*64+64) as four 16-wide slices -> four v8f accumulators.
// Per K-step (4 samples): A = one-hot(id, sample) 16x4 f32, B = emb 4x16 f32,
// C += A x B  via v_wmma_f32_16x16x4_f32 (f32 accumulation, no atomics).
// ---------------------------------------------------------------------------
__global__ void __launch_bounds__(256)
ea_agg_wmma_kernel(const float* __restrict__ emb,
                   const int*   __restrict__ labels,
                   const int*   __restrict__ order,
                   const int*   __restrict__ offsets,
                   const int*   __restrict__ counts,
                   const int*   __restrict__ rank,
                   float*       __restrict__ out) {
  const int seg_base = blockIdx.x * SEGS_PER_BLOCK;
  const int wave  = threadIdx.x >> 5;
  const int lane  = threadIdx.x & 31;
  const int m     = lane & 15;      // A row / C column index
  const int khalf = lane >> 4;      // 0 -> K{0,1}, 1 -> K{2,3}
  const int colb  = wave * 64;

  const int sbeg = offsets[seg_base];
  const int send = offsets[seg_base + SEGS_PER_BLOCK];
  const int cnt  = send - sbeg;

  v8f c0 = {}, c1 = {}, c2 = {}, c3 = {};

  for (int s = 0; s < cnt; s += 4) {      // uniform per block -> EXEC all 1s
    const int k0 = khalf * 2;
    const bool v0 = (s + k0)     < cnt;
    const bool v1 = (s + k0 + 1) < cnt;
    const int i0 = order[v0 ? (sbeg + s + k0)     : sbeg];
    const int i1 = order[v1 ? (sbeg + s + k0 + 1) : sbeg];
    const int l0 = labels[i0];
    const int l1 = labels[i1];

    v2f a;
    a.x = (v0 && (l0 == seg_base + m)) ? 1.0f : 0.0f;
    a.y = (v1 && (l1 == seg_base + m)) ? 1.0f : 0.0f;

    const float* r0 = emb + (size_t)i0 * D + colb;
    const float* r1 = emb + (size_t)i1 * D + colb;
    __builtin_prefetch(r0 + D, 0, 1);     // global_prefetch_b8 on next rows
    __builtin_prefetch(r1 + D, 0, 1);

    v2f b0, b1, b2, b3;
    b0.x = r0[m];      b0.y = r1[m];
    b1.x = r0[16 + m]; b1.y = r1[16 + m];
    b2.x = r0[32 + m]; b2.y = r1[32 + m];
    b3.x = r0[48 + m]; b3.y = r1[48 + m];

    c0 = __builtin_amdgcn_wmma_f32_16x16x4_f32(false, a, false, b0, (short)0, c0, false, false);
    c1 = __builtin_amdgcn_wmma_f32_16x16x4_f32(false, a, false, b1, (short)0, c1, false, false);
    c2 = __builtin_amdgcn_wmma_f32_16x16x4_f32(false, a, false, b2, (short)0, c2, false, false);
    c3 = __builtin_amdgcn_wmma_f32_16x16x4_f32(false, a, false, b3, (short)0, c3, false, false);
  }

  // C layout: vgpr j, lanes 0-15: M=j; lanes 16-31: M=8+j; N = lane&15.
  #pragma unroll
  for (int j = 0; j < 8; ++j) {
    const int g = seg_base + j + 8 * khalf;
    const int cg = counts[g];
    if (cg > 0) {
      const float inv = 1.0f / (float)cg;
      float* dst = out + (size_t)rank[g] * D + colb + m;
      dst[0]  = c0[j] * inv;
      dst[16] = c1[j] * inv;
      dst[32] = c2[j] * inv;
      dst[48] = c3[j] * inv;
    }
  }
}

// ---------------------------------------------------------------------------
extern "C" void kernel_launch(void* const* d_in, const int* in_sizes, int n_in,
                              void* d_out, int out_size, void* d_ws, size_t ws_size,
                              hipStream_t stream) {
  const float* emb    = (const float*)d_in[0];
  const int*   labels = (const int*)d_in[1];
  const int n = in_sizes[1];
  float* out = (float*)d_out;

  // workspace layout (ints): counts[8192] | offsets[8193] | cursor[8192]
  //                        | rank[8192]   | order[n]
  int* counts  = (int*)d_ws;
  int* offsets = counts + NUM_IDS;
  int* cursor  = offsets + NUM_IDS + 1;
  int* rank    = cursor + NUM_IDS;
  int* order   = rank + NUM_IDS;

  const int TOT = AGG_ELEMS + NUM_IDS;
  ea_init_kernel<<<(TOT + 255) / 256, 256, 0, stream>>>(out, counts);
  ea_hist_kernel<<<(n + 255) / 256, 256, 0, stream>>>(labels, n, counts);
  ea_scan_kernel<<<1, 256, 0, stream>>>(counts, offsets, cursor, rank,
                                        out + AGG_ELEMS, n);
  ea_scatter_kernel<<<(n + 255) / 256, 256, 0, stream>>>(labels, n, cursor, order);
  ea_agg_wmma_kernel<<<NUM_IDS / SEGS_PER_BLOCK, 256, 0, stream>>>(
      emb, labels, order, offsets, counts, rank, out);
}